// AtomCollection_81492709474775
// MI455X (gfx1250) — compile-verified
//
#include <hip/hip_runtime.h>
#include <math.h>

// ---------------------------------------------------------------------------
// AtomCollection for MI455X (gfx1250).
// Exact-math simplifications vs reference:
//  * softmax dropped (monotonic; straight-through logits == 1 identically)
//  * value conv evaluated only at the 512 selected points per batch
//  * conv_transpose done as sparse scatter-add of unit-norm atoms
// GEMMs (w_out 512x32, choice 512x512) use the CDNA5 f32 matrix pipe
// (V_WMMA_F32_16X16X4_F32): f32 in / f32 acc keeps top-k ordering faithful.
// choice GEMM: 64-column blocks, K chunked 4x128, double-buffered LDS panels
// filled with GLOBAL_LOAD_ASYNC_TO_LDS_B128 (ASYNCcnt pipeline) so HBM fetch
// of the next panel overlaps the WMMA stream of the current one.
// Workspace use: ~371 MB (fm 64MB, h0/h1 32MB, cbuf 268MB, ua 1MB, ctrl).
// ---------------------------------------------------------------------------

#define DEV __device__ __forceinline__

namespace {
constexpr int B_SZ  = 4;
constexpr int T_SZ  = 32768;
constexpr int FB_CH = 128;
constexpr int FB_K  = 128;
constexpr int MD    = 32;
constexpr int NA    = 512;
constexpr int KATOM = 512;
constexpr int KEEP  = 512;

// choice GEMM tiling
constexpr int CN  = 64;   // columns per block
constexpr int CKC = 128;  // K-chunk rows staged per LDS buffer
constexpr int CNCHUNK = NA / CKC; // 4

// workspace layout (float offsets)
constexpr size_t FM_OFF   = 0;                                   // B*128*T
constexpr size_t H0_OFF   = FM_OFF + (size_t)B_SZ * FB_CH * T_SZ;
constexpr size_t H1_OFF   = H0_OFF + (size_t)B_SZ * MD * T_SZ;
constexpr size_t CB_OFF   = H1_OFF + (size_t)B_SZ * MD * T_SZ;   // B*512*T
constexpr size_t UA_OFF   = CB_OFF + (size_t)B_SZ * NA * T_SZ;   // 512*512
constexpr size_t VL_OFF   = UA_OFF + (size_t)NA * KATOM;         // B*KEEP
constexpr size_t CTRL_OFF = VL_OFF + (size_t)B_SZ * KEEP;        // uints

// ctrl region (uint offsets)
constexpr int HIST_U   = 0;     // 4*256
constexpr int PREFIX_U = 1024;  // 4
constexpr int RANK_U   = 1028;  // 4
constexpr int CNT_U    = 1032;  // 4
constexpr int IDX_U    = 1036;  // 4*512
} // namespace

typedef __attribute__((ext_vector_type(2))) float v2f;
typedef __attribute__((ext_vector_type(8))) float v8f;

// D(16x16,f32) = A(16x4,f32) x B(4x16,f32) + C  -- wave32, 8-arg form.
DEV v8f wmma_f32_16x16x4(v2f a, v2f b, v8f c) {
#if __has_builtin(__builtin_amdgcn_wmma_f32_16x16x4_f32)
  return __builtin_amdgcn_wmma_f32_16x16x4_f32(
      /*neg_a=*/false, a, /*neg_b=*/false, b,
      /*c_mod=*/(short)0, c, /*reuse_a=*/false, /*reuse_b=*/false);
#else
  c[0] += a[0] * b[0] + a[1] * b[1]; // placeholder; primary path expected
  return c;
#endif
}

DEV unsigned ordered_u32(float f) {
  unsigned s = __float_as_uint(f);
  return s ^ ((s >> 31) ? 0xFFFFFFFFu : 0x80000000u);
}

// Issue one 16-byte async global->LDS copy (ASYNCcnt-tracked, GV mode).
// LDS byte address = low 32 bits of the generic pointer (aperture rules).
DEV void async_copy_b128(float* lds_dst, const float* gsrc) {
  unsigned lds = (unsigned)(uintptr_t)lds_dst;
  asm volatile("global_load_async_to_lds_b128 %0, %1, off"
               :: "v"(lds), "v"(gsrc)
               : "memory");
}
DEV void async_wait0() {
  asm volatile("s_wait_asynccnt 0x0" ::: "memory");
}

// ------------------------------- utility ----------------------------------

__global__ void zero_kernel(float* __restrict__ p, size_t n) {
  for (size_t i = (size_t)blockIdx.x * blockDim.x + threadIdx.x; i < n;
       i += (size_t)gridDim.x * blockDim.x)
    p[i] = 0.0f;
}

__global__ void ctrl_init_kernel(unsigned* __restrict__ ctrl) {
  int i = blockIdx.x * 256 + threadIdx.x;
  if (i < 1024) ctrl[HIST_U + i] = 0u;
  if (i < B_SZ) {
    ctrl[PREFIX_U + i] = 0u;
    ctrl[RANK_U + i]   = (unsigned)KEEP;
    ctrl[CNT_U + i]    = 0u;
  }
  if (i < B_SZ * KEEP) ctrl[IDX_U + i] = 0xFFFFFFFFu; // invalid marker
}

__global__ void atoms_norm_kernel(const float* __restrict__ atoms,
                                  float* __restrict__ ua) {
  __shared__ float red[256];
  const int a = blockIdx.x, tid = threadIdx.x;
  float s = 0.0f;
  for (int j = tid; j < KATOM; j += 256) {
    float v = atoms[(size_t)a * KATOM + j];
    s += v * v;
  }
  red[tid] = s;
  __syncthreads();
  for (int off = 128; off > 0; off >>= 1) {
    if (tid < off) red[tid] += red[tid + off];
    __syncthreads();
  }
  __shared__ float inv;
  if (tid == 0) inv = 1.0f / (sqrtf(red[0]) + 1e-12f);
  __syncthreads();
  for (int j = tid; j < KATOM; j += 256)
    ua[(size_t)a * KATOM + j] = atoms[(size_t)a * KATOM + j] * inv;
}

// --------------------------- front-end convs ------------------------------

// fm[b,ch,t] = sum_k fb_w[ch,k] * x[b, t+k-64]   ('same' pad (64,63))
__global__ void fbconv_kernel(const float* __restrict__ x,
                              const float* __restrict__ fw,
                              float* __restrict__ fm) {
  __shared__ float wl[64 * FB_K];   // 32 KB (two channel halves)
  __shared__ float xs[256 + FB_K];  // input tile + halo
  const int b = blockIdx.y, t0 = blockIdx.x * 256, tid = threadIdx.x;
  for (int i = tid; i < 256 + FB_K; i += 256) {
    int xi = t0 + i - FB_K / 2;
    xs[i] = (xi >= 0 && xi < T_SZ) ? x[(size_t)b * T_SZ + xi] : 0.0f;
  }
  for (int half = 0; half < 2; ++half) {
    __syncthreads();
    for (int i = tid; i < 64 * FB_K; i += 256)
      wl[i] = fw[(size_t)half * 64 * FB_K + i];
    __syncthreads();
    for (int ch = 0; ch < 64; ++ch) {
      float acc = 0.0f;
#pragma unroll 8
      for (int k = 0; k < FB_K; ++k) acc += wl[ch * FB_K + k] * xs[tid + k];
      fm[((size_t)b * FB_CH + half * 64 + ch) * T_SZ + t0 + tid] = acc;
    }
  }
}

// h0[b,m,t] = b_in[m] + sum_c w_in[m,c] * fm[b,c,t]
__global__ void win_kernel(const float* __restrict__ fm,
                           const float* __restrict__ W,
                           const float* __restrict__ bias,
                           float* __restrict__ h0) {
  __shared__ float wl[MD * FB_CH];
  __shared__ float bl[MD];
  const int b = blockIdx.y, t = blockIdx.x * 256 + threadIdx.x;
  for (int i = threadIdx.x; i < MD * FB_CH; i += 256) wl[i] = W[i];
  if (threadIdx.x < MD) bl[threadIdx.x] = bias[threadIdx.x];
  __syncthreads();
  float acc[MD];
#pragma unroll
  for (int m = 0; m < MD; ++m) acc[m] = bl[m];
  for (int c = 0; c < FB_CH; ++c) {
    float xv = fm[((size_t)b * FB_CH + c) * T_SZ + t];
#pragma unroll
    for (int m = 0; m < MD; ++m) acc[m] += wl[m * FB_CH + c] * xv;
  }
#pragma unroll
  for (int m = 0; m < MD; ++m) h0[((size_t)b * MD + m) * T_SZ + t] = acc[m];
}

// fused dilated 3-tap conv + 1x1 + residual + leaky_relu(0.2)
__global__ void dblock_kernel(const float* __restrict__ hin,
                              float* __restrict__ hout,
                              const float* __restrict__ dw,
                              const float* __restrict__ db,
                              const float* __restrict__ pw,
                              const float* __restrict__ pb, int dil) {
  __shared__ float dwl[MD * MD * 3];
  __shared__ float pwl[MD * MD];
  __shared__ float dbl[MD], pbl[MD];
  const int b = blockIdx.y, t = blockIdx.x * 256 + threadIdx.x;
  for (int i = threadIdx.x; i < MD * MD * 3; i += 256) dwl[i] = dw[i];
  for (int i = threadIdx.x; i < MD * MD; i += 256) pwl[i] = pw[i];
  if (threadIdx.x < MD) { dbl[threadIdx.x] = db[threadIdx.x]; pbl[threadIdx.x] = pb[threadIdx.x]; }
  __syncthreads();

  float tmp[MD];
#pragma unroll
  for (int m = 0; m < MD; ++m) tmp[m] = dbl[m];
  const int tm = t - dil, tp = t + dil;
  for (int c = 0; c < MD; ++c) {
    const size_t base = ((size_t)b * MD + c) * T_SZ;
    float x0 = (tm >= 0)   ? hin[base + tm] : 0.0f;
    float x1 = hin[base + t];
    float x2 = (tp < T_SZ) ? hin[base + tp] : 0.0f;
#pragma unroll
    for (int m = 0; m < MD; ++m) {
      const float* wp = &dwl[(m * MD + c) * 3];
      tmp[m] += wp[0] * x0 + wp[1] * x1 + wp[2] * x2;
    }
  }
#pragma unroll
  for (int m = 0; m < MD; ++m) {
    float z = pbl[m];
#pragma unroll
    for (int c = 0; c < MD; ++c) z += pwl[m * MD + c] * tmp[c];
    float r = z + hin[((size_t)b * MD + m) * T_SZ + t];
    hout[((size_t)b * MD + m) * T_SZ + t] = r > 0.0f ? r : 0.2f * r;
  }
}

// ------------------------------ WMMA GEMMs --------------------------------
// Fragment layout (16x16x4 f32):
//  A: lane l: m = l&15, k = k0 + (l>>4)*2  -> {A[m][k], A[m][k+1]}
//  B: lane l: n = l&15, k = k0 + (l>>4)*2  -> {B[k][n], B[k+1][n]}
//  C/D: VGPR v, lane l: M = v + (l>>4)*8, N = l&15

// full[b,o,t] = b_out[o] + sum_m W[o,m]*h[b,m,t]   (M=512, K=32, N=B*T)
__global__ void wout_wmma_kernel(const float* __restrict__ h,
                                 const float* __restrict__ W,
                                 const float* __restrict__ bias,
                                 float* __restrict__ full) {
  __shared__ float hs[MD * 16]; // 32 rows x 16 cols
  const int b = blockIdx.y, t0 = blockIdx.x * 16, tid = threadIdx.x;
  for (int i = tid; i < MD * 16; i += 256) {
    int kk = i >> 4, n = i & 15;
    hs[i] = h[((size_t)b * MD + kk) * T_SZ + t0 + n];
  }
  __syncthreads();
  const int wave = tid >> 5, lane = tid & 31;
  const int lhalf = lane >> 4, l15 = lane & 15;
  const int m0w = wave * 64;

  v8f acc[4];
#pragma unroll
  for (int mt = 0; mt < 4; ++mt)
#pragma unroll
    for (int v = 0; v < 8; ++v)
      acc[mt][v] = bias[m0w + mt * 16 + v + lhalf * 8];

#pragma unroll
  for (int k0 = 0; k0 < MD; k0 += 4) {
    const int kb = k0 + lhalf * 2;
    v2f bfrag;
    bfrag.x = hs[kb * 16 + l15];
    bfrag.y = hs[(kb + 1) * 16 + l15];
#pragma unroll
    for (int mt = 0; mt < 4; ++mt) {
      const int m = m0w + mt * 16 + l15;
      v2f afrag = *(const v2f*)(W + (size_t)m * MD + kb);
      acc[mt] = wmma_f32_16x16x4(afrag, bfrag, acc[mt]);
    }
  }
#pragma unroll
  for (int mt = 0; mt < 4; ++mt)
#pragma unroll
    for (int v = 0; v < 8; ++v) {
      const int M = m0w + mt * 16 + v + lhalf * 8;
      full[((size_t)b * NA + M) * T_SZ + t0 + l15] = acc[mt][v];
    }
}

// cbuf[b,o,t] = choice_b[o] + sum_a W[o,a]*full[b,a,t]  (M=512, K=512)
// Block: 64 columns, all 512 rows; 8 waves x (4 M-tiles x 4 N-tiles).
// B-panels (128 K-rows x 64 cols = 32KB) double-buffered in LDS, filled with
// async global->LDS b128 copies so the next panel's HBM fetch overlaps WMMA.
__global__ void choice_wmma_kernel(const float* __restrict__ full,
                                   const float* __restrict__ W,
                                   const float* __restrict__ bias,
                                   float* __restrict__ cbuf) {
  __shared__ float fs[2][CKC * CN]; // 2 x 32 KB
  const int b = blockIdx.y, t0 = blockIdx.x * CN, tid = threadIdx.x;
  const float* colbase = full + (size_t)b * NA * T_SZ + t0;

  // stage K-chunk `kc` into buffer `buf` (each thread: 8 x 16B, no divergence)
  auto stage = [&](float* buf, int kc) {
    for (int i = tid; i < CKC * (CN / 4); i += 256) {
      const int r = i >> 4, n4 = (i & 15) * 4;
      async_copy_b128(buf + r * CN + n4,
                      colbase + ((size_t)(kc * CKC + r)) * T_SZ + n4);
    }
  };

  stage(fs[0], 0);

  const int wave = tid >> 5, lane = tid & 31;
  const int lhalf = lane >> 4, l15 = lane & 15;
  const int m0w = wave * 64;

  v8f acc[4][4];
#pragma unroll
  for (int mt = 0; mt < 4; ++mt)
#pragma unroll
    for (int nt = 0; nt < 4; ++nt)
#pragma unroll
      for (int v = 0; v < 8; ++v)
        acc[mt][nt][v] = bias[m0w + mt * 16 + v + lhalf * 8];

  for (int c = 0; c < CNCHUNK; ++c) {
    async_wait0();
    __syncthreads();              // panel c resident; prior reads retired
    if (c + 1 < CNCHUNK) stage(fs[(c + 1) & 1], c + 1);
    const float* buf = fs[c & 1];
    const int kcb = c * CKC;

    for (int k0 = 0; k0 < CKC; k0 += 4) {
      const int kb = k0 + lhalf * 2;
      v2f bfrag[4];
#pragma unroll
      for (int nt = 0; nt < 4; ++nt) {
        bfrag[nt].x = buf[kb * CN + nt * 16 + l15];
        bfrag[nt].y = buf[(kb + 1) * CN + nt * 16 + l15];
      }
#pragma unroll
      for (int mt = 0; mt < 4; ++mt) {
        const int m = m0w + mt * 16 + l15;
        v2f afrag = *(const v2f*)(W + (size_t)m * NA + kcb + kb);
#pragma unroll
        for (int nt = 0; nt < 4; ++nt)
          acc[mt][nt] = wmma_f32_16x16x4(afrag, bfrag[nt], acc[mt][nt]);
      }
    }
  }

#pragma unroll
  for (int mt = 0; mt < 4; ++mt)
#pragma unroll
    for (int nt = 0; nt < 4; ++nt)
#pragma unroll
      for (int v = 0; v < 8; ++v) {
        const int M = m0w + mt * 16 + v + lhalf * 8;
        cbuf[((size_t)b * NA + M) * T_SZ + t0 + nt * 16 + l15] = acc[mt][nt][v];
      }
}

// --------------------------- top-512 (radix select) -----------------------

__global__ void hist_kernel(const float* __restrict__ cbuf,
                            unsigned* __restrict__ ctrl, int round) {
  __shared__ unsigned lh[256];
  const int b = blockIdx.y;
  lh[threadIdx.x] = 0u;
  __syncthreads();
  const unsigned prefix = ctrl[PREFIX_U + b];
  const int shift = 24 - 8 * round;
  const unsigned himask = (round == 0) ? 0u : (0xFFFFFFFFu << (shift + 8));
  const float* p = cbuf + (size_t)b * NA * T_SZ;
  const size_t n = (size_t)NA * T_SZ;
  for (size_t i = (size_t)blockIdx.x * 256 + threadIdx.x; i < n;
       i += (size_t)gridDim.x * 256) {
    unsigned u = ordered_u32(p[i]);
    if ((u & himask) == (prefix & himask))
      atomicAdd(&lh[(u >> shift) & 255u], 1u);
  }
  __syncthreads();
  if (lh[threadIdx.x]) atomicAdd(&ctrl[HIST_U + b * 256 + threadIdx.x], lh[threadIdx.x]);
}

__global__ void scan_kernel(unsigned* __restrict__ ctrl, int round) {
  const int b = threadIdx.x;
  if (b < B_SZ) {
    const int shift = 24 - 8 * round;
    unsigned r = ctrl[RANK_U + b];
    unsigned cum = 0;
    int sel = 0;
    for (int bb = 255; bb >= 0; --bb) {
      unsigned h = ctrl[HIST_U + b * 256 + bb];
      if (cum + h >= r) { sel = bb; break; }
      cum += h;
    }
    ctrl[PREFIX_U + b] |= ((unsigned)sel) << shift;
    ctrl[RANK_U + b] = r - cum;
    for (int bb = 0; bb < 256; ++bb) ctrl[HIST_U + b * 256 + bb] = 0u;
  }
}

// pass 0: strictly above threshold (count <= 511 by construction)
// pass 1: ties at the threshold fill the remaining slots
__global__ void collect_kernel(const float* __restrict__ cbuf,
                               unsigned* __restrict__ ctrl, int tiePass) {
  const int b = blockIdx.y;
  const unsigned thr = ctrl[PREFIX_U + b];
  const float* p = cbuf + (size_t)b * NA * T_SZ;
  const size_t n = (size_t)NA * T_SZ;
  for (size_t i = (size_t)blockIdx.x * 256 + threadIdx.x; i < n;
       i += (size_t)gridDim.x * 256) {
    unsigned u = ordered_u32(p[i]);
    bool take = tiePass ? (u == thr) : (u > thr);
    if (take) {
      unsigned slot = atomicAdd(&ctrl[CNT_U + b], 1u);
      if (slot < (unsigned)KEEP) ctrl[IDX_U + b * KEEP + slot] = (unsigned)i;
    }
  }
}

// ----------------------- sparse values + reconstruction -------------------

__global__ void value_scatter_kernel(const float* __restrict__ fullp,
                                     const float* __restrict__ vw,
                                     const float* __restrict__ vb,
                                     const unsigned* __restrict__ ctrl,
                                     float* __restrict__ vlist,
                                     float* __restrict__ sparse) {
  __shared__ float red[256];
  const int b = blockIdx.y, s = blockIdx.x, tid = threadIdx.x;
  const unsigned idx = ctrl[IDX_U + b * KEEP + s];
  if (idx >= (unsigned)(NA * T_SZ)) return; // slot never filled
  const int o = (int)(idx >> 15);    // T = 2^15
  const int t = (int)(idx & (T_SZ - 1));
  float part = 0.0f;
  for (int a = tid; a < NA; a += 256)
    part += vw[(size_t)o * NA + a] * fullp[((size_t)b * NA + a) * T_SZ + t];
  red[tid] = part;
  __syncthreads();
  for (int off = 128; off > 0; off >>= 1) {
    if (tid < off) red[tid] += red[tid + off];
    __syncthreads();
  }
  if (tid == 0) {
    float v = red[0] + vb[o];   // straight-through logits == 1
    sparse[((size_t)b * NA + o) * T_SZ + t] = v;
    vlist[b * KEEP + s] = v;
  }
}

// rec[b, tau + j - 256] += val * ua[o, j]
__global__ void rec_kernel(const unsigned* __restrict__ ctrl,
                           const float* __restrict__ vlist,
                           const float* __restrict__ ua,
                           float* __restrict__ rec) {
  const int b = blockIdx.y, s = blockIdx.x, tid = threadIdx.x;
  const unsigned idx = ctrl[IDX_U + b * KEEP + s];
  if (idx >= (unsigned)(NA * T_SZ)) return;
  const float val = vlist[b * KEEP + s];
  const int o = (int)(idx >> 15);
  const int tau = (int)(idx & (T_SZ - 1));
  for (int j = tid; j < KATOM; j += 256) {
    int t = tau + j - 256;
    if (t >= 0 && t < T_SZ - 1)
      atomicAdd(&rec[(size_t)b * (T_SZ - 1) + t], val * ua[(size_t)o * KATOM + j]);
  }
}

// ------------------------------- launcher ---------------------------------

extern "C" void kernel_launch(void* const* d_in, const int* in_sizes, int n_in,
                              void* d_out, int out_size, void* d_ws, size_t ws_size,
                              hipStream_t stream) {
  (void)in_sizes; (void)n_in; (void)out_size; (void)ws_size;
  const float* x     = (const float*)d_in[0];
  const float* fb_w  = (const float*)d_in[1];
  const float* w_in  = (const float*)d_in[2];
  const float* b_in  = (const float*)d_in[3];
  const float* d1w   = (const float*)d_in[4];
  const float* d1b   = (const float*)d_in[5];
  const float* p1w   = (const float*)d_in[6];
  const float* p1b   = (const float*)d_in[7];
  const float* d2w   = (const float*)d_in[8];
  const float* d2b   = (const float*)d_in[9];
  const float* p2w   = (const float*)d_in[10];
  const float* p2b   = (const float*)d_in[11];
  const float* w_out = (const float*)d_in[12];
  const float* b_out = (const float*)d_in[13];
  const float* ch_w  = (const float*)d_in[14];
  const float* ch_b  = (const float*)d_in[15];
  const float* va_w  = (const float*)d_in[16];
  const float* va_b  = (const float*)d_in[17];
  const float* atoms = (const float*)d_in[18];

  float* wsf   = (float*)d_ws;
  float* fm    = wsf + FM_OFF;
  float* h0    = wsf + H0_OFF;
  float* h1    = wsf + H1_OFF;
  float* cb    = wsf + CB_OFF;
  float* ua    = wsf + UA_OFF;
  float* vlist = wsf + VL_OFF;
  unsigned* ctrl = (unsigned*)(wsf + CTRL_OFF);

  float* out_sparse = (float*)d_out;                                // [B,512,T]
  float* out_rec    = out_sparse + (size_t)B_SZ * NA * T_SZ;        // [B,1,T-1]
  float* out_full   = out_rec + (size_t)B_SZ * (T_SZ - 1);          // [B,512,T]

  ctrl_init_kernel<<<8, 256, 0, stream>>>(ctrl);
  zero_kernel<<<4096, 256, 0, stream>>>(
      out_sparse, (size_t)B_SZ * NA * T_SZ + (size_t)B_SZ * (T_SZ - 1));
  atoms_norm_kernel<<<NA, 256, 0, stream>>>(atoms, ua);

  dim3 gT(T_SZ / 256, B_SZ);
  fbconv_kernel<<<gT, 256, 0, stream>>>(x, fb_w, fm);
  win_kernel<<<gT, 256, 0, stream>>>(fm, w_in, b_in, h0);
  dblock_kernel<<<gT, 256, 0, stream>>>(h0, h1, d1w, d1b, p1w, p1b, 1);
  dblock_kernel<<<gT, 256, 0, stream>>>(h1, h0, d2w, d2b, p2w, p2b, 3);

  wout_wmma_kernel<<<dim3(T_SZ / 16, B_SZ), 256, 0, stream>>>(
      h0, w_out, b_out, out_full);
  choice_wmma_kernel<<<dim3(T_SZ / CN, B_SZ), 256, 0, stream>>>(
      out_full, ch_w, ch_b, cb);

  for (int r = 0; r < 4; ++r) {
    hist_kernel<<<dim3(1024, B_SZ), 256, 0, stream>>>(cb, ctrl, r);
    scan_kernel<<<1, 32, 0, stream>>>(ctrl, r);
  }
  collect_kernel<<<dim3(1024, B_SZ), 256, 0, stream>>>(cb, ctrl, 0);
  collect_kernel<<<dim3(1024, B_SZ), 256, 0, stream>>>(cb, ctrl, 1);

  value_scatter_kernel<<<dim3(KEEP, B_SZ), 256, 0, stream>>>(
      out_full, va_w, va_b, ctrl, vlist, out_sparse);
  rec_kernel<<<dim3(KEEP, B_SZ), 256, 0, stream>>>(ctrl, vlist, ua, out_rec);
}